// PatternsOfThinkingBlock_30623116820924
// MI455X (gfx1250) — compile-verified
//
#include <hip/hip_runtime.h>
#include <math.h>

// ---------------------------------------------------------------------------
// Reference:  x:[2,16,2048,2048] f32, W:[2048,2048] f32, b:[2048] f32
//   idx      = argmax(x, -1)                      [B,H,S]  (first occurrence)
//   gathered = x[..., idx]                        [B,H,S]  (row max)
//   y        = gelu_exact(W @ gathered + b)       [B,H,S]
//   out      = x with out[b,h,s,idx[b,h,s]] = y[b,h,s]
//
// MI455X plan (memory bound, ~1.04 GB traffic @ 23.3 TB/s ≈ 45us floor):
//   K1: stream-copy x->out + per-row argmax (wave32 per row, b128 ld/st)
//   K2: tiny GEMM (M=32,N=2048,K=2048) with V_WMMA_F32_16X16X4_F32
//       (fp32 path: matches reference precision), fused GELU + scatter.
// ---------------------------------------------------------------------------

typedef float v2f __attribute__((ext_vector_type(2)));
typedef float v4f __attribute__((ext_vector_type(4)));
typedef float v8f __attribute__((ext_vector_type(8)));

#define SEQ   2048
#define NROW  65536   /* 2*16*2048 rows of length SEQ */
#define NBH   32      /* 2*16 batch*head */

// ---------------------------------------------------------------------------
// Kernel 1: one wave32 per row. Copy row x->out (nontemporal b128), track
// per-lane (max, first index), then xor-shuffle wave reduction with
// first-occurrence tie-break (equal value -> smaller index wins).
// ---------------------------------------------------------------------------
__global__ __launch_bounds__(256) void argmax_copy_kernel(
    const float* __restrict__ x, float* __restrict__ out,
    float* __restrict__ g, int* __restrict__ idxArr)
{
    const int wave = threadIdx.x >> 5;
    const int lane = threadIdx.x & 31;
    const long row = (long)blockIdx.x * 8 + wave;

    const float* __restrict__ xr = x + row * SEQ;
    float* __restrict__ orow     = out + row * SEQ;

    float best = -INFINITY;
    int   bidx = 0;

    // 2048 floats / wave: lane handles 16 x float4; index = i*128 + lane*4 + j
    // (per-lane traversal order is strictly increasing, so strict '>' keeps
    //  the first occurrence within a lane)
#pragma unroll
    for (int i = 0; i < 16; ++i) {
        const int e = i * 128 + lane * 4;
        v4f v = __builtin_nontemporal_load((const v4f*)(xr + e));
        __builtin_nontemporal_store(v, (v4f*)(orow + e));
#pragma unroll
        for (int j = 0; j < 4; ++j) {
            float f = v[j];
            if (f > best) { best = f; bidx = e + j; }
        }
    }

    // wave32 xor reduction; tie -> smaller index (first occurrence)
#pragma unroll
    for (int off = 16; off >= 1; off >>= 1) {
        float ov = __shfl_xor(best, off, 32);
        int   oi = __shfl_xor(bidx, off, 32);
        if (ov > best || (ov == best && oi < bidx)) { best = ov; bidx = oi; }
    }

    if (lane == 0) {
        g[row]      = best;
        idxArr[row] = bidx;
    }
}

// ---------------------------------------------------------------------------
// Kernel 2: D[m,n] = sum_k A[m,k]*B[k,n],  m=bh (32), n=t (2048), k=s (2048)
//   A[m,k] = g[m*SEQ + k]            (gathered)
//   B[k,n] = W[n*SEQ + k]            (W row-major [t,s])
// One wave per 16x16 D tile; K in steps of 4 via V_WMMA_F32_16X16X4_F32.
//
// VGPR layouts (CDNA5 ISA 7.12.2):
//   A 16x4 : VGPR v, lanes 0-15 -> (M=lane, K=v); lanes 16-31 -> (M=lane-16, K=v+2)
//   B 4x16 : assumed mirror of A: VGPR v, half h -> (K=v+2h, N=lane&15)
//   C/D    : VGPR r, lanes 0-15 -> (M=r, N=lane); lanes 16-31 -> (M=r+8, N=lane-16)
// Each lane therefore streams contiguous float2s from one row of g and one
// row of W -> W's 16 MB read exactly once, L2-resident for the 2nd M tile.
// Epilogue: +bias, exact GELU (erf), scatter into out at argmax positions.
// ---------------------------------------------------------------------------
__global__ __launch_bounds__(32) void wmma_gemm_gelu_scatter_kernel(
    const float* __restrict__ g, const float* __restrict__ W,
    const float* __restrict__ bias, const int* __restrict__ idxArr,
    float* __restrict__ out)
{
    const int lane = threadIdx.x & 31;
    const int n0   = blockIdx.x * 16;   // t tile
    const int m0   = blockIdx.y * 16;   // bh tile
    const int half = lane >> 4;         // 0: K pair {0,1}, 1: K pair {2,3}
    const int l16  = lane & 15;
    const int koff = half * 2;

    const float* __restrict__ gRow = g + (long)(m0 + l16) * SEQ;  // A row (M=l16)
    const float* __restrict__ wRow = W + (long)(n0 + l16) * SEQ;  // B col source (N=l16)

    v8f c = {0.f, 0.f, 0.f, 0.f, 0.f, 0.f, 0.f, 0.f};

    for (int kb = 0; kb < SEQ; kb += 4) {
        v2f a = *(const v2f*)(gRow + kb + koff);   // K = kb+koff, kb+koff+1
        v2f b = *(const v2f*)(wRow + kb + koff);
        // (neg_a, A, neg_b, B, c_mod, C, reuse_a, reuse_b)
        c = __builtin_amdgcn_wmma_f32_16x16x4_f32(
                false, a, false, b, (short)0, c, false, false);
    }

    const int   t  = n0 + l16;
    const float bv = bias[t];

#pragma unroll
    for (int r = 0; r < 8; ++r) {
        const int bh = m0 + r + half * 8;          // D row (M) for this element
        float v = c[r] + bv;
        // exact GELU: 0.5*v*(1+erf(v/sqrt(2)))  (torch/jax approximate=False)
        float y = 0.5f * v * (1.0f + erff(v * 0.70710678118654752440f));
        const int id = idxArr[bh * SEQ + t];
        out[((long)bh * SEQ + t) * SEQ + id] = y;
    }
}

// ---------------------------------------------------------------------------
extern "C" void kernel_launch(void* const* d_in, const int* in_sizes, int n_in,
                              void* d_out, int out_size, void* d_ws, size_t ws_size,
                              hipStream_t stream)
{
    const float* x    = (const float*)d_in[0];  // [2,16,2048,2048]
    const float* W    = (const float*)d_in[1];  // [2048,2048]
    const float* bias = (const float*)d_in[2];  // [2048]
    float*       out  = (float*)d_out;          // [2,16,2048,2048]

    // workspace: gathered[NROW] f32 | idx[NROW] i32  (512 KB total)
    float* g      = (float*)d_ws;
    int*   idxArr = (int*)((char*)d_ws + (size_t)NROW * sizeof(float));

    // K1: 8 waves/block, 1 wave per row -> 8192 blocks
    argmax_copy_kernel<<<NROW / 8, 256, 0, stream>>>(x, out, g, idxArr);

    // K2: 128 N-tiles x 2 M-tiles, one wave each
    wmma_gemm_gelu_scatter_kernel<<<dim3(SEQ / 16, NBH / 16), 32, 0, stream>>>(
        g, W, bias, idxArr, out);
}